// GCNIDS_60344290509380
// MI455X (gfx1250) — compile-verified
//
#include <hip/hip_runtime.h>
#include <hip/hip_bf16.h>
#include <stdint.h>

typedef __attribute__((ext_vector_type(2))) float v2f;
typedef __attribute__((ext_vector_type(8))) float v8f;
typedef __attribute__((ext_vector_type(4))) unsigned int u32x4;
typedef __attribute__((ext_vector_type(8))) int i32x8;
typedef __attribute__((ext_vector_type(4))) int i32x4;

#define IN_DIM  128
#define HID_DIM 256
#define OUT_DIM 128

#if defined(__has_builtin)
#if __has_builtin(__builtin_amdgcn_tensor_load_to_lds) && __has_builtin(__builtin_amdgcn_s_wait_tensorcnt)
#define HAVE_TDM 1
#endif
#endif

// Native CDNA5 fp32 global atomic add (non-returning, STOREcnt-tracked).
__device__ __forceinline__ void atomic_add_f32_global(float* p, float v) {
  asm volatile("global_atomic_add_f32 %0, %1, off"
               :: "v"((unsigned long long)(uintptr_t)p), "v"(v)
               : "memory");
}

// ---------------- degree / norm ----------------
__global__ void deg_init_k(int* __restrict__ cnt, int n) {
  int i = blockIdx.x * blockDim.x + threadIdx.x;
  if (i < n) cnt[i] = 1;  // self loop
}

__global__ void deg_count_k(const int* __restrict__ dst, int* __restrict__ cnt, int e) {
  int i = blockIdx.x * blockDim.x + threadIdx.x;
  if (i < e) atomicAdd(&cnt[dst[i]], 1);
}

__global__ void dinv_k(const int* __restrict__ cnt, float* __restrict__ dinv, int n) {
  int i = blockIdx.x * blockDim.x + threadIdx.x;
  if (i < n) dinv[i] = rsqrtf((float)cnt[i]);
}

// ---------------- fp32 WMMA GEMM: C[M,NC] = (relu?)A[M,K] * B[K,NC] ----------------
// Block = 8 waves, covers 16 rows x 128 cols of C. The block's B panel (K x 128)
// is DMA'd into LDS by the Tensor Data Mover, then each wave runs the K-loop with
// V_WMMA_F32_16X16X4_F32, A fragments from global (b64), B fragments from LDS.
template <bool RELU_IN, int K, int NC>
__global__ void gemm_wmma_k(const float* __restrict__ A, const float* __restrict__ B,
                            float* __restrict__ C) {
  extern __shared__ float bsh[];              // K*128 floats
  const int tid  = threadIdx.x;
  const int lane = tid & 31;
  const int wv   = tid >> 5;                  // 8 waves
  const int n0blk = blockIdx.y * 128;

#if HAVE_TDM
  if (wv == 0) {
    // Tensor DMA descriptor: 2D tensor B[K][NC], tile K x 128 at column n0blk.
    unsigned long long gaddr = (unsigned long long)(uintptr_t)(B + n0blk);
    unsigned int ldsoff = (unsigned int)(uintptr_t)bsh;  // low 32 bits = LDS byte offset
    u32x4 g0;
    g0[0] = 1u;                                           // count=1 (valid user D#)
    g0[1] = ldsoff;                                       // lds_addr
    g0[2] = (unsigned int)(gaddr & 0xFFFFFFFFu);          // global_addr[31:0]
    g0[3] = (unsigned int)((gaddr >> 32) & 0x1FFFFFFu)    // global_addr[56:32]
            | (2u << 30);                                 // type=2 ("image")
    i32x8 g1;
    g1[0] = (int)(2u << 16);                              // data_size=2 -> 4 bytes
    g1[1] = (int)(((unsigned)NC & 0xFFFFu) << 16);        // tensor_dim0[15:0]
    g1[2] = (int)((((unsigned)NC >> 16) & 0xFFFFu)        // tensor_dim0[31:16]
            | (((unsigned)K & 0xFFFFu) << 16));           // tensor_dim1[15:0]
    g1[3] = (int)((((unsigned)K >> 16) & 0xFFFFu)         // tensor_dim1[31:16]
            | (128u << 16));                              // tile_dim0 = 128
    g1[4] = (int)((unsigned)K & 0xFFFFu);                 // tile_dim1 = K, tile_dim2 = 0
    g1[5] = (int)(unsigned)NC;                            // tensor_dim0_stride[31:0]
    g1[6] = 0;                                            // stride hi / dim1_stride lo
    g1[7] = 0;
    i32x4 gz4 = {0, 0, 0, 0};                             // 2D: groups 2/3 unused
    i32x8 gz8 = {0, 0, 0, 0, 0, 0, 0, 0};                 // unused trailing descriptor words
    __builtin_amdgcn_tensor_load_to_lds(g0, g1, gz4, gz4, gz8, 0);
    __builtin_amdgcn_s_wait_tensorcnt(0);
  }
  __syncthreads();
#else
  // cooperative global->LDS fallback
  for (int idx = tid; idx < (K * 128) / 4; idx += 256) {
    int row = idx >> 5;                       // 32 float4 groups per 128-wide row
    int c4  = idx & 31;
    float4 v = *(const float4*)(B + (size_t)row * NC + n0blk + c4 * 4);
    *(float4*)(bsh + row * 128 + c4 * 4) = v;
  }
  __syncthreads();
#endif

  const int half = lane >> 4;                 // 0: K{0,1}/M0-7 ; 1: K{2,3}/M8-15
  const int l15  = lane & 15;
  const size_t row = (size_t)blockIdx.x * 16 + l15;
  const float* Ap = A + row * K + (half << 1);
  const float* bp = bsh + wv * 16 + l15;      // this wave's column in the LDS panel

  v8f c = {};
  #pragma unroll 8
  for (int k0 = 0; k0 < K; k0 += 4) {
    v2f a = *(const v2f*)(Ap + k0);           // b64 load, immediate offset
    if (RELU_IN) { a[0] = fmaxf(a[0], 0.f); a[1] = fmaxf(a[1], 0.f); }
    const int kr = k0 + (half << 1);
    v2f b;
    b[0] = bp[kr * 128];                      // ds_load_b32
    b[1] = bp[(kr + 1) * 128];
    c = __builtin_amdgcn_wmma_f32_16x16x4_f32(false, a, false, b,
                                              (short)0, c, false, false);
  }

  float* Cp = C + ((size_t)blockIdx.x * 16 + (half << 3)) * NC + n0blk + wv * 16 + l15;
  #pragma unroll
  for (int v = 0; v < 8; ++v) Cp[(size_t)v * NC] = c[v];
}

// ---------------- aggregation ----------------
// agg[i][:] = bias[:] + dinv[i]^2 * h[i][:]   (self-loop term, atomic-free)
template <int GSHIFT>
__global__ void agg_init_k(const float* __restrict__ h, const float* __restrict__ bias,
                           const float* __restrict__ dinv, float* __restrict__ agg, int n) {
  const int D4 = 1 << GSHIFT;
  int t = blockIdx.x * blockDim.x + threadIdx.x;
  int i = t >> GSHIFT;
  int g = t & (D4 - 1);
  if (i >= n) return;
  float s = dinv[i]; s *= s;
  float4 hv = ((const float4*)h)[(size_t)i * D4 + g];
  float4 bv = ((const float4*)bias)[g];
  float4 o;
  o.x = bv.x + s * hv.x; o.y = bv.y + s * hv.y;
  o.z = bv.z + s * hv.z; o.w = bv.w + s * hv.w;
  ((float4*)agg)[(size_t)i * D4 + g] = o;
}

// agg[dst][:] += dinv[src]*dinv[dst] * h[src][:]  (fp32 global atomics, L2-resident)
template <int GSHIFT>
__global__ void scatter_k(const int* __restrict__ src, const int* __restrict__ dst,
                          const float* __restrict__ dinv, const float* __restrict__ h,
                          float* __restrict__ agg, int e) {
  const int D4 = 1 << GSHIFT;
  int t = blockIdx.x * blockDim.x + threadIdx.x;
  int ei = t >> GSHIFT;
  int g  = t & (D4 - 1);
  if (ei >= e) return;
  int s = src[ei], d = dst[ei];
  float nrm = dinv[s] * dinv[d];
  float4 hv = ((const float4*)h)[(size_t)s * D4 + g];
  float* p = agg + ((size_t)d * D4 + g) * 4;
  atomic_add_f32_global(p + 0, nrm * hv.x);
  atomic_add_f32_global(p + 1, nrm * hv.y);
  atomic_add_f32_global(p + 2, nrm * hv.z);
  atomic_add_f32_global(p + 3, nrm * hv.w);
}

// ---------------- head: out[i] = relu(h[i]) . Wout + bout ----------------
__global__ void final_k(const float* __restrict__ h, const float* __restrict__ w,
                        const float* __restrict__ bout, float* __restrict__ out, int n) {
  int gt = blockIdx.x * blockDim.x + threadIdx.x;
  int node = gt >> 5;                         // one wave32 per node: 128 dims = 32 float4
  int lane = gt & 31;
  if (node >= n) return;
  float4 hv = ((const float4*)h)[(size_t)node * 32 + lane];
  float4 wv = ((const float4*)w)[lane];
  float s = fmaxf(hv.x, 0.f) * wv.x + fmaxf(hv.y, 0.f) * wv.y +
            fmaxf(hv.z, 0.f) * wv.z + fmaxf(hv.w, 0.f) * wv.w;
  #pragma unroll
  for (int off = 16; off > 0; off >>= 1) s += __shfl_xor(s, off, 32);
  if (lane == 0) out[node] = s + bout[0];
}

extern "C" void kernel_launch(void* const* d_in, const int* in_sizes, int n_in,
                              void* d_out, int out_size, void* d_ws, size_t ws_size,
                              hipStream_t stream) {
  const float* x    = (const float*)d_in[0];
  const int*   ei   = (const int*)d_in[1];
  const float* W1   = (const float*)d_in[2];
  const float* b1   = (const float*)d_in[3];
  const float* W2   = (const float*)d_in[4];
  const float* b2   = (const float*)d_in[5];
  const float* Wout = (const float*)d_in[6];
  const float* bout = (const float*)d_in[7];
  float* out = (float*)d_out;

  const int N = in_sizes[0] / IN_DIM;   // 100000 (divisible by 16)
  const int E = in_sizes[1] / 2;        // 1600000
  const int* src = ei;
  const int* dst = ei + E;

  // workspace layout: cnt | dinv | bufA (N*256) | bufB (N*256)
  char* ws = (char*)d_ws;
  size_t offDinv = ((size_t)N * 4 + 255) & ~(size_t)255;
  size_t offA    = (offDinv + (size_t)N * 4 + 255) & ~(size_t)255;
  size_t offB    = (offA + (size_t)N * HID_DIM * 4 + 255) & ~(size_t)255;
  int*   cnt  = (int*)ws;
  float* dinv = (float*)(ws + offDinv);
  float* bufA = (float*)(ws + offA);
  float* bufB = (float*)(ws + offB);

  const int B256 = 256;
  // 1) degrees + dinv
  deg_init_k<<<(N + B256 - 1) / B256, B256, 0, stream>>>(cnt, N);
  deg_count_k<<<(E + B256 - 1) / B256, B256, 0, stream>>>(dst, cnt, E);
  dinv_k<<<(N + B256 - 1) / B256, B256, 0, stream>>>(cnt, dinv, N);

  // 2) h1pre = x @ W1 -> bufA   (TDM-staged B panel, 64KB LDS)
  gemm_wmma_k<false, IN_DIM, HID_DIM>
      <<<dim3(N / 16, HID_DIM / 128), B256, IN_DIM * 128 * 4, stream>>>(x, W1, bufA);

  // 3) agg1 = b1 + dinv^2*h1pre + edge scatter -> bufB   (D=256)
  agg_init_k<6><<<(N * (HID_DIM / 4) + B256 - 1) / B256, B256, 0, stream>>>(
      bufA, b1, dinv, bufB, N);
  scatter_k<6><<<(E + 3) / 4, B256, 0, stream>>>(src, dst, dinv, bufA, bufB, E);

  // 4) h2pre = relu(agg1) @ W2 -> bufA   (128KB LDS panel)
  gemm_wmma_k<true, HID_DIM, OUT_DIM>
      <<<dim3(N / 16, OUT_DIM / 128), B256, HID_DIM * 128 * 4, stream>>>(bufB, W2, bufA);

  // 5) agg2 = b2 + dinv^2*h2pre + edge scatter -> bufB   (D=128)
  agg_init_k<5><<<(N * (OUT_DIM / 4) + B256 - 1) / B256, B256, 0, stream>>>(
      bufA, b2, dinv, bufB, N);
  scatter_k<5><<<(E + 7) / 8, B256, 0, stream>>>(src, dst, dinv, bufA, bufB, E);

  // 6) out = relu(agg2) @ Wout + bout
  final_k<<<((N * 32) + B256 - 1) / B256, B256, 0, stream>>>(bufB, Wout, bout, out, N);
}